// Top2Gate_43267500539993
// MI455X (gfx1250) — compile-verified
//
#include <hip/hip_runtime.h>
#include <hip/hip_bf16.h>

// Problem constants (match reference)
#define S_TOK 8192
#define MDIM  2048
#define NEXP  64
#define CAPN  256
static constexpr size_t SEC = (size_t)S_TOK * NEXP * CAPN;   // 134,217,728

typedef float v2f __attribute__((ext_vector_type(2)));
typedef float v8f __attribute__((ext_vector_type(8)));

// ---------------------------------------------------------------------------
// Kernel 1: logits[S,E] = x[S,M] @ wg[E,M]^T  via V_WMMA_F32_16X16X4_F32.
// One wave computes a 16-token x 64-expert tile (4 accumulators of 16x16).
// A layout (32-bit 16x4): lanes 0-15 -> M=lane, K={0,1}; lanes 16-31 -> M=lane-16, K={2,3}.
// B layout mirrored: lanes hold N=lane&15, K={kSel,kSel+1} where kSel = (lane>>4)*2.
// ---------------------------------------------------------------------------
__global__ __launch_bounds__(256) void gemm_wmma_kernel(const float* __restrict__ x,
                                                        const float* __restrict__ wg,
                                                        float* __restrict__ logits) {
  const int lane    = threadIdx.x & 31;
  const int wave    = threadIdx.x >> 5;
  const int rowBase = (blockIdx.x * 8 + wave) * 16;   // 64 blocks * 8 waves * 16 rows = 8192
  const int half    = lane >> 4;                      // 0 or 1
  const int sub     = lane & 15;
  const int kSel    = half * 2;

  const float* __restrict__ aP = x + (size_t)(rowBase + sub) * MDIM + kSel;
  const float* __restrict__ bP0 = wg + (size_t)(0 * 16 + sub) * MDIM + kSel;
  const float* __restrict__ bP1 = wg + (size_t)(1 * 16 + sub) * MDIM + kSel;
  const float* __restrict__ bP2 = wg + (size_t)(2 * 16 + sub) * MDIM + kSel;
  const float* __restrict__ bP3 = wg + (size_t)(3 * 16 + sub) * MDIM + kSel;

  v8f acc0 = {}, acc1 = {}, acc2 = {}, acc3 = {};

#pragma unroll 4
  for (int kk = 0; kk < MDIM; kk += 4) {
    v2f a  = *(const v2f*)(aP  + kk);
    v2f b0 = *(const v2f*)(bP0 + kk);
    v2f b1 = *(const v2f*)(bP1 + kk);
    v2f b2 = *(const v2f*)(bP2 + kk);
    v2f b3 = *(const v2f*)(bP3 + kk);
    acc0 = __builtin_amdgcn_wmma_f32_16x16x4_f32(false, a, false, b0, (short)0, acc0, false, false);
    acc1 = __builtin_amdgcn_wmma_f32_16x16x4_f32(false, a, false, b1, (short)0, acc1, false, false);
    acc2 = __builtin_amdgcn_wmma_f32_16x16x4_f32(false, a, false, b2, (short)0, acc2, false, false);
    acc3 = __builtin_amdgcn_wmma_f32_16x16x4_f32(false, a, false, b3, (short)0, acc3, false, false);
  }

  // C/D layout: VGPR i, lanes 0-15 -> (M=i, N=lane); lanes 16-31 -> (M=8+i, N=lane-16)
  float* outp = logits + (size_t)rowBase * NEXP;
#pragma unroll
  for (int i = 0; i < 8; ++i) {
    const int r = (i + 8 * half) * NEXP + sub;
    outp[r +  0] = acc0[i];
    outp[r + 16] = acc1[i];
    outp[r + 32] = acc2[i];
    outp[r + 48] = acc3[i];
  }
}

// ---------------------------------------------------------------------------
// Kernel 2: per-token gating. One wave per token, 2 experts per lane.
// ---------------------------------------------------------------------------
__global__ __launch_bounds__(256) void gate_kernel(const float* __restrict__ logits,
                                                   const float* __restrict__ gumbel,
                                                   float* __restrict__ gates,
                                                   int* __restrict__ idx1,
                                                   int* __restrict__ idx2,
                                                   float* __restrict__ g1v,
                                                   float* __restrict__ g2v) {
  const int token = blockIdx.x * 8 + (threadIdx.x >> 5);
  const int lane  = threadIdx.x & 31;

  const float2 lp = *(const float2*)(logits + (size_t)token * NEXP + lane * 2);

  // softmax over 64 experts
  float m = fmaxf(lp.x, lp.y);
#pragma unroll
  for (int off = 16; off > 0; off >>= 1) m = fmaxf(m, __shfl_xor(m, off, 32));
  const float e0 = expf(lp.x - m);
  const float e1 = expf(lp.y - m);
  float ssum = e0 + e1;
#pragma unroll
  for (int off = 16; off > 0; off >>= 1) ssum += __shfl_xor(ssum, off, 32);
  const float inv = 1.0f / ssum;
  const float gx = e0 * inv, gy = e1 * inv;
  *(float2*)(gates + (size_t)token * NEXP + lane * 2) = make_float2(gx, gy);

  // top-1 over logits (same argmax as gates); ties -> lower index like jnp.argmax
  float bv; int bi;
  if (lp.x >= lp.y) { bv = lp.x; bi = lane * 2; } else { bv = lp.y; bi = lane * 2 + 1; }
#pragma unroll
  for (int off = 16; off > 0; off >>= 1) {
    float ov = __shfl_xor(bv, off, 32);
    int   oi = __shfl_xor(bi, off, 32);
    if (ov > bv || (ov == bv && oi < bi)) { bv = ov; bi = oi; }
  }
  const int i1 = bi;
  const int sl1 = i1 >> 1;
  const float bgx1 = __shfl(gx, sl1, 32);
  const float bgy1 = __shfl(gy, sl1, 32);
  const float gate1 = (i1 & 1) ? bgy1 : bgx1;

  // top-2: argmax over gumbel-noised logits with top-1 masked to -inf
  const float2 gb = *(const float2*)(gumbel + (size_t)token * NEXP + lane * 2);
  const float NEGINF = -__builtin_inff();
  const float n0 = (lane * 2     == i1) ? NEGINF : lp.x + gb.x;
  const float n1 = (lane * 2 + 1 == i1) ? NEGINF : lp.y + gb.y;
  if (n0 >= n1) { bv = n0; bi = lane * 2; } else { bv = n1; bi = lane * 2 + 1; }
#pragma unroll
  for (int off = 16; off > 0; off >>= 1) {
    float ov = __shfl_xor(bv, off, 32);
    int   oi = __shfl_xor(bi, off, 32);
    if (ov > bv || (ov == bv && oi < bi)) { bv = ov; bi = oi; }
  }
  const int i2 = bi;
  const int sl2 = i2 >> 1;
  const float bgx2 = __shfl(gx, sl2, 32);
  const float bgy2 = __shfl(gy, sl2, 32);
  const float gate2 = (i2 & 1) ? bgy2 : bgx2;

  if (lane == 0) {
    idx1[token] = i1;
    idx2[token] = i2;
    g1v[token]  = gate1;
    g2v[token]  = gate2;
  }
}

// ---------------------------------------------------------------------------
// Kernel 3: per-expert sequential prefix scan over token order (cumsum of
// one-hot masks). One wave per expert; ballot/popc prefix within 32-chunks.
// ---------------------------------------------------------------------------
__global__ __launch_bounds__(32) void scan_kernel(const int* __restrict__ idx1,
                                                  const int* __restrict__ idx2,
                                                  int* __restrict__ loc1,
                                                  int* __restrict__ loc2,
                                                  float* __restrict__ cnt1) {
  const int e = blockIdx.x;
  const int lane = threadIdx.x;
  const unsigned lt = (1u << lane) - 1u;

  int run = 0;
  for (int base = 0; base < S_TOK; base += 32) {
    const int s = base + lane;
    const int mine = (idx1[s] == e) ? 1 : 0;
    const unsigned mask = (unsigned)__ballot(mine);
    if (mine) loc1[s] = run + __popc(mask & lt);
    run += __popc(mask);
  }
  const int total1 = run;
  if (lane == 0) cnt1[e] = (float)total1;   // ce numerator (pre-capacity-drop)

  int run2 = 0;
  for (int base = 0; base < S_TOK; base += 32) {
    const int s = base + lane;
    const int mine = (idx2[s] == e) ? 1 : 0;
    const unsigned mask = (unsigned)__ballot(mine);
    if (mine) loc2[s] = total1 + run2 + __popc(mask & lt);
    run2 += __popc(mask);
  }
}

// ---------------------------------------------------------------------------
// Kernel 4: me_sum[e] = sum_s gates[s,e]   (block per expert, no atomics)
// ---------------------------------------------------------------------------
__global__ __launch_bounds__(256) void me_kernel(const float* __restrict__ gates,
                                                 float* __restrict__ me_sum) {
  const int e = blockIdx.x;
  const int t = threadIdx.x;
  float sum = 0.0f;
  for (int s = t; s < S_TOK; s += 256) sum += gates[(size_t)s * NEXP + e];
  __shared__ float red[256];
  red[t] = sum;
  __syncthreads();
#pragma unroll
  for (int off = 128; off > 0; off >>= 1) {
    if (t < off) red[t] += red[t + off];
    __syncthreads();
  }
  if (t == 0) me_sum[e] = red[0];
}

// ---------------------------------------------------------------------------
// Kernel 5: zero-fill d_out (l_aux + combine + dispatch = 2*SEC+1 floats)
// with b128 stores; tail element handled by thread 0.
// ---------------------------------------------------------------------------
__global__ __launch_bounds__(256) void zero_kernel(float4* __restrict__ out4) {
  const size_t n4 = (2 * SEC) / 4;
  const size_t stride = (size_t)gridDim.x * blockDim.x;
  const float4 z = make_float4(0.f, 0.f, 0.f, 0.f);
  for (size_t i = (size_t)blockIdx.x * blockDim.x + threadIdx.x; i < n4; i += stride)
    out4[i] = z;
  if (blockIdx.x == 0 && threadIdx.x == 0)
    ((float*)out4)[2 * SEC] = 0.0f;    // the one leftover element
}

// ---------------------------------------------------------------------------
// Kernel 6: l_aux = E * sum_e(me_sum[e]*cnt1[e]) / S^2   -> d_out[0]
// ---------------------------------------------------------------------------
__global__ __launch_bounds__(64) void laux_kernel(const float* __restrict__ me_sum,
                                                  const float* __restrict__ cnt1,
                                                  float* __restrict__ out) {
  const int t = threadIdx.x;
  __shared__ float red[64];
  red[t] = me_sum[t] * cnt1[t];
  __syncthreads();
#pragma unroll
  for (int off = 32; off > 0; off >>= 1) {
    if (t < off) red[t] += red[t + off];
    __syncthreads();
  }
  if (t == 0)
    out[0] = red[0] * ((float)NEXP / ((float)S_TOK * (float)S_TOK));
}

// ---------------------------------------------------------------------------
// Kernel 7: scatter the <=2 nonzero entries per token into combine/dispatch.
// ---------------------------------------------------------------------------
__global__ __launch_bounds__(256) void scatter_kernel(const int* __restrict__ idx1,
                                                      const int* __restrict__ idx2,
                                                      const int* __restrict__ loc1,
                                                      const int* __restrict__ loc2,
                                                      const float* __restrict__ g1v,
                                                      const float* __restrict__ g2v,
                                                      float* __restrict__ out) {
  const int s = blockIdx.x * 256 + threadIdx.x;
  if (s >= S_TOK) return;
  const int i1 = idx1[s], i2 = idx2[s];
  const int l1 = loc1[s], l2 = loc2[s];
  const bool m1 = (l1 < CAPN);
  const bool m2 = (l2 < CAPN);
  float g1 = m1 ? g1v[s] : 0.0f;
  float g2 = m2 ? g2v[s] : 0.0f;
  const float denom = fmaxf(g1 + g2, 1.1920928955078125e-07f);   // f32 eps
  g1 /= denom;
  g2 /= denom;
  float* __restrict__ comb = out + 1;
  float* __restrict__ disp = out + 1 + SEC;
  if (m1 && g1 > 0.0f) {
    const size_t o = (size_t)s * (NEXP * CAPN) + (size_t)i1 * CAPN + l1;
    comb[o] = g1;
    disp[o] = 1.0f;
  }
  if (m2 && g2 > 0.0f) {
    const size_t o = (size_t)s * (NEXP * CAPN) + (size_t)i2 * CAPN + l2;
    comb[o] = g2;
    disp[o] = 1.0f;
  }
}

// ---------------------------------------------------------------------------
// Workspace layout (units of 4 bytes):
//   [0, S*E)                 logits
//   [S*E, 2*S*E)             gates
//   then int regions: idx1[S], idx2[S], loc1[S], loc2[S]
//   then float: g1v[S], g2v[S], cnt1[64], me_sum[64]
// total ~4.4 MB
// ---------------------------------------------------------------------------
extern "C" void kernel_launch(void* const* d_in, const int* in_sizes, int n_in,
                              void* d_out, int out_size, void* d_ws, size_t ws_size,
                              hipStream_t stream) {
  const float* x      = (const float*)d_in[0];   // [S, M]
  const float* wg     = (const float*)d_in[1];   // [E, M]
  const float* gumbel = (const float*)d_in[2];   // [S, E]
  float* out = (float*)d_out;

  char* ws = (char*)d_ws;
  const size_t SE = (size_t)S_TOK * NEXP;
  float* logits = (float*)ws;                       // S*E floats
  float* gates  = logits + SE;                      // S*E floats
  int*   idx1   = (int*)(gates + SE);               // S ints
  int*   idx2   = idx1 + S_TOK;
  int*   loc1   = idx2 + S_TOK;
  int*   loc2   = loc1 + S_TOK;
  float* g1v    = (float*)(loc2 + S_TOK);
  float* g2v    = g1v + S_TOK;
  float* cnt1   = g2v + S_TOK;                      // 64 floats
  float* me_sum = cnt1 + NEXP;                      // 64 floats

  // 1) logits = x @ wg^T     (WMMA f32, 64 blocks * 8 waves * 16 rows)
  gemm_wmma_kernel<<<S_TOK / (8 * 16), 256, 0, stream>>>(x, wg, logits);

  // 2) softmax + top1/top2 per token (wave per token)
  gate_kernel<<<S_TOK / 8, 256, 0, stream>>>(logits, gumbel, gates, idx1, idx2, g1v, g2v);

  // 3) per-expert capacity prefix scans (wave per expert)
  scan_kernel<<<NEXP, 32, 0, stream>>>(idx1, idx2, loc1, loc2, cnt1);

  // 4) column sums of gates for l_aux
  me_kernel<<<NEXP, 256, 0, stream>>>(gates, me_sum);

  // 5) zero the 1.07 GB output (must happen every call; harness doesn't re-poison)
  zero_kernel<<<8192, 256, 0, stream>>>((float4*)out);

  // 6) l_aux -> out[0] (after zero-fill)
  laux_kernel<<<1, 64, 0, stream>>>(me_sum, cnt1, out);

  // 7) sparse scatter of combine/dispatch entries
  scatter_kernel<<<(S_TOK + 255) / 256, 256, 0, stream>>>(idx1, idx2, loc1, loc2, g1v, g2v, out);
}